// MQANDecoder_69123203661890
// MI455X (gfx1250) — compile-verified
//
#include <hip/hip_runtime.h>
#include <hip/hip_bf16.h>
#include <math.h>

typedef __attribute__((ext_vector_type(2))) float v2f;
typedef __attribute__((ext_vector_type(8))) float v8f;

namespace {
constexpr int B_  = 128;   // batch
constexpr int T_  = 64;    // time steps
constexpr int S_  = 512;   // context length
constexpr int DH_ = 512;   // hidden
constexpr int G_  = 2048;  // 4*DH_
constexpr int LDS_STRIDE = 36;      // padded row stride (floats) for B tile
constexpr int KC = 32;              // K chunk staged per barrier
}

// ---------------------------------------------------------------------------
// fp32 WMMA (V_WMMA_F32_16X16X4_F32) accumulation of one A(16xK) @ W^T slice
// into four 16x16 accumulators covering a 16x64 output tile, with the 64xKC
// B slice staged in LDS once per block (8 waves = 8 M-tiles share it).
// A frag (32-bit 16x4): lanes0-15 row M=lm K={k,k+1}; lanes16-31 K={k+2,k+3}
// B frag (4x16): lanes0-15 row K=k (N=lm); lanes16-31 row K=k+2
// ---------------------------------------------------------------------------
__device__ __forceinline__ void gemm_src_lds(float* __restrict__ bs,
                                             const float* __restrict__ A, long lda,
                                             const float* __restrict__ W, long ldw, int K,
                                             int m0, int n0, int tid,
                                             v8f& c0, v8f& c1, v8f& c2, v8f& c3)
{
  const int lane = tid & 31;
  const int lm = lane & 15, half = lane >> 4;
  const int sn = tid >> 2;            // 0..63 : N row staged by this thread
  const int kp = (tid & 3) * 8;       // 0,8,16,24 : K sub-range staged
  const float* arow = A + (long)(m0 + lm) * lda + 2 * half;

  for (int k0 = 0; k0 < K; k0 += KC) {
    __syncthreads();
    {
      const float4* src = (const float4*)(W + (long)(n0 + sn) * ldw + k0 + kp);
      float4 w0 = src[0];
      float4 w1 = src[1];
      float* dst = bs + sn * LDS_STRIDE + kp;
      dst[0] = w0.x; dst[1] = w0.y; dst[2] = w0.z; dst[3] = w0.w;
      dst[4] = w1.x; dst[5] = w1.y; dst[6] = w1.z; dst[7] = w1.w;
    }
    __syncthreads();
    const float* ar = arow + k0;
    const float* b0p = bs + lm * LDS_STRIDE + 2 * half;
    const float* b1p = b0p + 16 * LDS_STRIDE;
    const float* b2p = b0p + 32 * LDS_STRIDE;
    const float* b3p = b0p + 48 * LDS_STRIDE;
#pragma unroll
    for (int kk = 0; kk < KC; kk += 4) {
      float2 av  = *(const float2*)(ar + kk);
      float2 b0v = *(const float2*)(b0p + kk);
      float2 b1v = *(const float2*)(b1p + kk);
      float2 b2v = *(const float2*)(b2p + kk);
      float2 b3v = *(const float2*)(b3p + kk);
      v2f a  = {av.x,  av.y};
      v2f b0 = {b0v.x, b0v.y};
      v2f b1 = {b1v.x, b1v.y};
      v2f b2 = {b2v.x, b2v.y};
      v2f b3 = {b3v.x, b3v.y};
      c0 = __builtin_amdgcn_wmma_f32_16x16x4_f32(false, a, false, b0, (short)0, c0, false, false);
      c1 = __builtin_amdgcn_wmma_f32_16x16x4_f32(false, a, false, b1, (short)0, c1, false, false);
      c2 = __builtin_amdgcn_wmma_f32_16x16x4_f32(false, a, false, b2, (short)0, c2, false, false);
      c3 = __builtin_amdgcn_wmma_f32_16x16x4_f32(false, a, false, b3, (short)0, c3, false, false);
    }
  }
}

// D(128 x N) = sum of up to three A_i @ W_i^T.
// grid: N/64 blocks of 256 threads; wave w handles M-tile w, block handles one N-group.
__global__ __launch_bounds__(256) void mqan_gemm_gates(
    const float* __restrict__ A0, long lda0, const float* __restrict__ W0, long ldw0, int K0,
    const float* __restrict__ A1, long lda1, const float* __restrict__ W1, long ldw1, int K1,
    const float* __restrict__ A2, long lda2, const float* __restrict__ W2, long ldw2, int K2,
    float* __restrict__ D, long ldd)
{
  __shared__ float bs[64 * LDS_STRIDE];
  const int tid = threadIdx.x;
  const int lane = tid & 31;
  const int lm = lane & 15, half = lane >> 4;
  const int m0 = (tid >> 5) * 16;       // 8 waves = 8 M-tiles
  const int n0 = blockIdx.x * 64;
  v8f c0 = {}, c1 = {}, c2 = {}, c3 = {};
  gemm_src_lds(bs, A0, lda0, W0, ldw0, K0, m0, n0, tid, c0, c1, c2, c3);
  gemm_src_lds(bs, A1, lda1, W1, ldw1, K1, m0, n0, tid, c0, c1, c2, c3);
  if (K2 > 0) gemm_src_lds(bs, A2, lda2, W2, ldw2, K2, m0, n0, tid, c0, c1, c2, c3);
  const int r0 = m0 + 8 * half;
#pragma unroll
  for (int v = 0; v < 8; ++v) {
    float* dr = D + (long)(r0 + v) * ldd + n0 + lm;
    dr[0]  = c0[v];
    dr[16] = c1[v];
    dr[32] = c2[v];
    dr[48] = c3[v];
  }
}

// out = tanh(dec @ Wout_l^T + align @ Wout_r^T); fan the result out to
// ctx_out (recurrent feed), context_outputs, and switch_inputs[.,512:1024].
__global__ __launch_bounds__(256) void mqan_gemm_out(
    const float* __restrict__ A0, long lda0, const float* __restrict__ W0, long ldw0, int K0,
    const float* __restrict__ A1, long lda1, const float* __restrict__ W1, long ldw1, int K1,
    float* __restrict__ ctxout,
    float* __restrict__ outp, long ldo,
    float* __restrict__ swp,  long lsw)
{
  __shared__ float bs[64 * LDS_STRIDE];
  const int tid = threadIdx.x;
  const int lane = tid & 31;
  const int lm = lane & 15, half = lane >> 4;
  const int m0 = (tid >> 5) * 16;
  const int n0 = blockIdx.x * 64;
  v8f c0 = {}, c1 = {}, c2 = {}, c3 = {};
  gemm_src_lds(bs, A0, lda0, W0, ldw0, K0, m0, n0, tid, c0, c1, c2, c3);
  gemm_src_lds(bs, A1, lda1, W1, ldw1, K1, m0, n0, tid, c0, c1, c2, c3);
  const int r0 = m0 + 8 * half;
#pragma unroll
  for (int v = 0; v < 8; ++v) {
    const long r = r0 + v;
#pragma unroll
    for (int j = 0; j < 4; ++j) {
      float val = (j == 0) ? c0[v] : (j == 1) ? c1[v] : (j == 2) ? c2[v] : c3[v];
      float th = tanhf(val);
      const int cidx = n0 + j * 16 + lm;
      ctxout[r * DH_ + cidx] = th;
      outp[r * ldo + cidx]   = th;
      swp[r * lsw + cidx]    = th;
    }
  }
}

__device__ __forceinline__ float sigf(float x) { return 1.0f / (1.0f + expf(-x)); }

// LSTM cell pointwise update (PyTorch gate order i,f,g,o). Optionally scatters
// dec and x_t slices into switch_inputs (layer 1 only).
__global__ __launch_bounds__(256) void mqan_lstm_cell(
    const float* __restrict__ gates, const float* __restrict__ b_ih,
    const float* __restrict__ b_hh,
    float* __restrict__ h, float* __restrict__ c,
    float* __restrict__ sw_dec, const float* __restrict__ x_t,
    float* __restrict__ sw_x, long lsw, long lx)
{
  const int idx = blockIdx.x * blockDim.x + threadIdx.x;  // B*DH threads
  const int b = idx >> 9, u = idx & 511;
  const long gb = (long)b * G_;
  float gi = gates[gb + u]          + b_ih[u]          + b_hh[u];
  float gf = gates[gb + 512 + u]    + b_ih[512 + u]    + b_hh[512 + u];
  float gg = gates[gb + 1024 + u]   + b_ih[1024 + u]   + b_hh[1024 + u];
  float go = gates[gb + 1536 + u]   + b_ih[1536 + u]   + b_hh[1536 + u];
  float cn = sigf(gf) * c[idx] + sigf(gi) * tanhf(gg);
  float hn = sigf(go) * tanhf(cn);
  c[idx] = cn;
  h[idx] = hn;
  if (sw_dec != nullptr) {
    sw_dec[(long)b * lsw + u] = hn;
    sw_x[(long)b * lsw + u]   = x_t[(long)b * lx + u];
  }
}

// Dot attention: one block per batch element. scores -> masked softmax -> align.
__global__ __launch_bounds__(256) void mqan_attn(
    const float* __restrict__ dec, const float* __restrict__ ctx,
    const unsigned char* __restrict__ mask,
    float* __restrict__ alignb, float* __restrict__ attn_out, long lat)
{
  __shared__ float sdec[DH_];
  __shared__ float sattn[S_];
  __shared__ float sred[256];
  const int b = blockIdx.x, tid = threadIdx.x;
  const float* cb = ctx + (long)b * S_ * DH_;
  sdec[tid]       = dec[(long)b * DH_ + tid];
  sdec[tid + 256] = dec[(long)b * DH_ + tid + 256];
  __syncthreads();

  float sc[2];
#pragma unroll
  for (int j = 0; j < 2; ++j) {
    const int s = tid + j * 256;
    const float* cp = cb + (long)s * DH_;
    float acc = 0.f;
#pragma unroll 8
    for (int d = 0; d < DH_; ++d) acc += sdec[d] * cp[d];
    if (mask[(long)b * S_ + s]) acc = -3.402823466e38f;
    sc[j] = acc;
  }
  // max reduce
  sred[tid] = fmaxf(sc[0], sc[1]);
  __syncthreads();
  for (int off = 128; off > 0; off >>= 1) {
    if (tid < off) sred[tid] = fmaxf(sred[tid], sred[tid + off]);
    __syncthreads();
  }
  const float mx = sred[0];
  __syncthreads();
  float e0 = expf(sc[0] - mx), e1 = expf(sc[1] - mx);
  sred[tid] = e0 + e1;
  __syncthreads();
  for (int off = 128; off > 0; off >>= 1) {
    if (tid < off) sred[tid] += sred[tid + off];
    __syncthreads();
  }
  const float inv = 1.0f / sred[0];
  const float a0 = e0 * inv, a1 = e1 * inv;
  sattn[tid] = a0; sattn[tid + 256] = a1;
  attn_out[(long)b * lat + tid]       = a0;
  attn_out[(long)b * lat + tid + 256] = a1;
  __syncthreads();
  // align[d] = sum_s attn[s]*ctx[b,s,d]  (s-outer => coalesced over d)
  float al0 = 0.f, al1 = 0.f;
  for (int s = 0; s < S_; ++s) {
    const float w = sattn[s];
    const float* cp = cb + (long)s * DH_;
    al0 += w * cp[tid];
    al1 += w * cp[tid + 256];
  }
  alignb[(long)b * DH_ + tid]       = al0;
  alignb[(long)b * DH_ + tid + 256] = al1;
}

// copy initial h0/c0 into working buffers, zero ctx_out feed
__global__ void mqan_init(const float* __restrict__ h0, const float* __restrict__ c0,
                          float* __restrict__ h0b, float* __restrict__ h1b,
                          float* __restrict__ c0b, float* __restrict__ c1b,
                          float* __restrict__ ctxout)
{
  const int i = blockIdx.x * blockDim.x + threadIdx.x;  // B*DH
  h0b[i] = h0[i];
  h1b[i] = h0[B_ * DH_ + i];
  c0b[i] = c0[i];
  c1b[i] = c0[B_ * DH_ + i];
  ctxout[i] = 0.0f;
}

__global__ void mqan_final(const float* __restrict__ h0b, const float* __restrict__ h1b,
                           const float* __restrict__ c0b, const float* __restrict__ c1b,
                           float* __restrict__ hfin, float* __restrict__ cfin)
{
  const int i = blockIdx.x * blockDim.x + threadIdx.x;  // B*DH
  hfin[i] = h0b[i];
  hfin[B_ * DH_ + i] = h1b[i];
  cfin[i] = c0b[i];
  cfin[B_ * DH_ + i] = c1b[i];
}

extern "C" void kernel_launch(void* const* d_in, const int* in_sizes, int n_in,
                              void* d_out, int out_size, void* d_ws, size_t ws_size,
                              hipStream_t stream) {
  const float* x    = (const float*)d_in[0];   // (B,T,512)
  const float* ctx  = (const float*)d_in[1];   // (B,S,512)
  const unsigned char* mask = (const unsigned char*)d_in[2];  // (B,S) bool
  const float* h0   = (const float*)d_in[3];   // (2,B,512)
  const float* c0   = (const float*)d_in[4];
  const float* Wih0 = (const float*)d_in[5];   // (2048,1024)
  const float* Whh0 = (const float*)d_in[6];   // (2048,512)
  const float* bih0 = (const float*)d_in[7];
  const float* bhh0 = (const float*)d_in[8];
  const float* Wih1 = (const float*)d_in[9];   // (2048,512)
  const float* Whh1 = (const float*)d_in[10];  // (2048,512)
  const float* bih1 = (const float*)d_in[11];
  const float* bhh1 = (const float*)d_in[12];
  const float* Wout = (const float*)d_in[13];  // (512,1024)

  // output layout (flat, return order)
  float* out  = (float*)d_out;
  float* outs = out;                                   // (B,T,512)
  float* sw   = outs + (size_t)B_ * T_ * DH_;          // (B,T,1536)
  float* attn = sw + (size_t)B_ * T_ * 3 * DH_;        // (B,T,512)
  float* hfin = attn + (size_t)B_ * T_ * DH_;          // (2,B,512)
  float* cfin = hfin + (size_t)2 * B_ * DH_;

  // workspace layout
  float* w = (float*)d_ws;
  float* ctxout = w; w += (size_t)B_ * DH_;
  float* h0b = w;    w += (size_t)B_ * DH_;
  float* h1b = w;    w += (size_t)B_ * DH_;
  float* c0b = w;    w += (size_t)B_ * DH_;
  float* c1b = w;    w += (size_t)B_ * DH_;
  float* alignb = w; w += (size_t)B_ * DH_;
  float* gates0 = w; w += (size_t)B_ * G_;
  float* gates1 = w; w += (size_t)B_ * G_;

  mqan_init<<<(B_ * DH_) / 256, 256, 0, stream>>>(h0, c0, h0b, h1b, c0b, c1b, ctxout);

  const long ldx = (long)T_ * DH_;
  for (int t = 0; t < T_; ++t) {
    // layer 0 gates: x_t @ Wih0[:, :512]^T + ctx_out @ Wih0[:, 512:]^T + h0 @ Whh0^T
    mqan_gemm_gates<<<G_ / 64, 256, 0, stream>>>(
        x + (size_t)t * DH_, ldx, Wih0, 1024, 512,
        ctxout, DH_, Wih0 + 512, 1024, 512,
        h0b, DH_, Whh0, 512, 512,
        gates0, G_);
    mqan_lstm_cell<<<(B_ * DH_) / 256, 256, 0, stream>>>(
        gates0, bih0, bhh0, h0b, c0b, nullptr, nullptr, nullptr, 0, 0);
    // layer 1 gates: nh0 @ Wih1^T + h1 @ Whh1^T
    mqan_gemm_gates<<<G_ / 64, 256, 0, stream>>>(
        h0b, DH_, Wih1, 512, 512,
        h1b, DH_, Whh1, 512, 512,
        nullptr, 0, nullptr, 0, 0,
        gates1, G_);
    mqan_lstm_cell<<<(B_ * DH_) / 256, 256, 0, stream>>>(
        gates1, bih1, bhh1, h1b, c1b,
        sw + (size_t)t * 3 * DH_,            // switch dec slice
        x + (size_t)t * DH_,                 // x_t
        sw + (size_t)t * 3 * DH_ + 2 * DH_,  // switch x slice
        (long)T_ * 3 * DH_, ldx);
    // attention (dec = h1b) -> alignb + attentions
    mqan_attn<<<B_, 256, 0, stream>>>(
        h1b, ctx, mask, alignb, attn + (size_t)t * DH_, (long)T_ * DH_);
    // out = tanh([dec, align] @ Wout^T) -> ctxout / context_outputs / switch mid slice
    mqan_gemm_out<<<DH_ / 64, 256, 0, stream>>>(
        h1b, DH_, Wout, 1024, 512,
        alignb, DH_, Wout + 512, 1024, 512,
        ctxout,
        outs + (size_t)t * DH_, (long)T_ * DH_,
        sw + (size_t)t * 3 * DH_ + DH_, (long)T_ * 3 * DH_);
  }

  mqan_final<<<(B_ * DH_) / 256, 256, 0, stream>>>(h0b, h1b, c0b, c1b, hfin, cfin);
}